// ROIHead_532575944979
// MI455X (gfx1250) — compile-verified
//
#include <hip/hip_runtime.h>
#include <hip/hip_bf16.h>
#include <math.h>

typedef float v2f __attribute__((ext_vector_type(2)));
typedef float v8f __attribute__((ext_vector_type(8)));

#define POOLP 7
#define NCH 256
#define FH 32
#define FW 32
#define NROI 512
#define DIN 12544           // 256*7*7
#define DHID 1024
#define NCLS 21
#define NBOX 10240          // 512 * 20 (classes 1..20)
#define NSORT 16384
#define TOPK 100
#define BBOX_CLIP 4.135166556742356f

// ---------------------------------------------------------------------------
// 1) ROI max-pool (torchvision roi_pool semantics, matches reference exactly)
// ---------------------------------------------------------------------------
__global__ void roi_pool_kernel(const float* __restrict__ feat,
                                const float* __restrict__ props,
                                float* __restrict__ X) {
  int id = blockIdx.x * blockDim.x + threadIdx.x;
  const int total = NROI * NCH * POOLP * POOLP;
  if (id >= total) return;
  int n   = id / (NCH * 49);
  int rem = id % (NCH * 49);
  int c   = rem / 49;
  int bin = rem % 49;
  int ph = bin / POOLP, pw = bin % POOLP;

  const float* r4 = props + n * 4;
  int x1 = (int)floorf(r4[0] * 0.0625f + 0.5f);
  int y1 = (int)floorf(r4[1] * 0.0625f + 0.5f);
  int x2 = (int)floorf(r4[2] * 0.0625f + 0.5f);
  int y2 = (int)floorf(r4[3] * 0.0625f + 0.5f);
  float rw = (float)max(x2 - x1 + 1, 1);
  float rh = (float)max(y2 - y1 + 1, 1);
  float bw = rw / (float)POOLP;
  float bh = rh / (float)POOLP;

  int ws = min(max((int)floorf((float)pw * bw) + x1, 0), FW);
  int we = min(max((int)ceilf ((float)(pw + 1) * bw) + x1, 0), FW);
  int hs = min(max((int)floorf((float)ph * bh) + y1, 0), FH);
  int he = min(max((int)ceilf ((float)(ph + 1) * bh) + y1, 0), FH);

  float val = -INFINITY;
  const float* fc = feat + (size_t)c * FH * FW;
  for (int y = hs; y < he; ++y)
    for (int x = ws; x < we; ++x)
      val = fmaxf(val, fc[y * FW + x]);
  if (he <= hs || we <= ws) val = 0.0f;   // empty window -> -inf -> 0 in ref
  X[(size_t)n * DIN + c * 49 + bin] = val;
}

// ---------------------------------------------------------------------------
// 2) LDS-tiled f32 WMMA GEMM:  C[M,Nb] = relu?( A[M,K] * B[Nb,K]^T + bias )
//    Block: 128 threads = 4 waves; block tile 64(M) x 64(N); K tiles of 32.
//    Each wave: 16(M) x 64(N) via 4 accumulators of V_WMMA_F32_16X16X4_F32.
// ---------------------------------------------------------------------------
#define TILE_K 32
#define LDS_STRIDE 36

__global__ __launch_bounds__(128) void gemm_wmma_f32(
    const float* __restrict__ A, const float* __restrict__ B,
    const float* __restrict__ bias, float* __restrict__ C,
    int K, int Nb, int relu) {
  __shared__ float As[64][LDS_STRIDE];
  __shared__ float Bs[64][LDS_STRIDE];

  const int tid  = threadIdx.x;
  const int lane = tid & 31;
  const int wave = tid >> 5;            // 0..3 -> M sub-tile
  const int m0 = blockIdx.x * 64;
  const int n0 = blockIdx.y * 64;
  const int halfsel = lane >> 4;        // 0 -> K pair {0,1}, 1 -> {2,3}
  const int l16 = lane & 15;

  v8f acc[4] = {};

  for (int k0 = 0; k0 < K; k0 += TILE_K) {
    // Stage A tile 64x32 (float4 coalesced)
    #pragma unroll
    for (int p = 0; p < 4; ++p) {
      int f = tid + p * 128;            // float4 index 0..511
      int row = f >> 3;
      int col4 = (f & 7) << 2;
      float4 v = *(const float4*)(A + (size_t)(m0 + row) * K + k0 + col4);
      *(float4*)&As[row][col4] = v;
    }
    // Stage B tile 64x32 (rows of W, zero-pad past Nb)
    #pragma unroll
    for (int p = 0; p < 4; ++p) {
      int f = tid + p * 128;
      int row = f >> 3;
      int col4 = (f & 7) << 2;
      int n = n0 + row;
      float4 v = make_float4(0.f, 0.f, 0.f, 0.f);
      if (n < Nb) v = *(const float4*)(B + (size_t)n * K + k0 + col4);
      *(float4*)&Bs[row][col4] = v;
    }
    // Prefetch next K tile while we compute (global_prefetch_b8)
    if (k0 + TILE_K < K) {
      __builtin_prefetch(A + (size_t)(m0 + (tid & 63)) * K + k0 + TILE_K, 0, 3);
      int pn = min(n0 + (tid & 63), Nb - 1);
      __builtin_prefetch(B + (size_t)pn * K + k0 + TILE_K, 0, 3);
    }
    __syncthreads();

    #pragma unroll
    for (int kk = 0; kk < TILE_K; kk += 4) {
      // A fragment: lane holds A[m=l16][kk + 2*halfsel + {0,1}]
      v2f a = *(const v2f*)&As[wave * 16 + l16][kk + halfsel * 2];
      #pragma unroll
      for (int t = 0; t < 4; ++t) {
        // B fragment: lane holds B^T[kk + 2*halfsel + {0,1}][n=l16]
        v2f b = *(const v2f*)&Bs[t * 16 + l16][kk + halfsel * 2];
        acc[t] = __builtin_amdgcn_wmma_f32_16x16x4_f32(
            false, a, false, b, (short)0, acc[t], false, false);
      }
    }
    __syncthreads();
  }

  // Epilogue: D layout — VGPR r: lanes0-15 M=r, lanes16-31 M=8+r; N = l16
  #pragma unroll
  for (int t = 0; t < 4; ++t) {
    int n = n0 + t * 16 + l16;
    if (n >= Nb) continue;
    float bv = bias[n];
    #pragma unroll
    for (int r = 0; r < 8; ++r) {
      int m = m0 + wave * 16 + r + halfsel * 8;
      float v = acc[t][r] + bv;
      if (relu) v = fmaxf(v, 0.0f);
      C[(size_t)m * Nb + n] = v;
    }
  }
}

// ---------------------------------------------------------------------------
// 3) Softmax over 21 classes per ROI
// ---------------------------------------------------------------------------
__global__ void softmax_kernel(const float* __restrict__ cls,
                               float* __restrict__ probs) {
  int n = blockIdx.x * blockDim.x + threadIdx.x;
  if (n >= NROI) return;
  const float* row = cls + (size_t)n * NCLS;
  float mx = row[0];
  for (int c = 1; c < NCLS; ++c) mx = fmaxf(mx, row[c]);
  float e[NCLS];
  float sum = 0.0f;
  for (int c = 0; c < NCLS; ++c) { e[c] = expf(row[c] - mx); sum += e[c]; }
  float inv = 1.0f / sum;
  for (int c = 0; c < NCLS; ++c) probs[(size_t)n * NCLS + c] = e[c] * inv;
}

// ---------------------------------------------------------------------------
// 4) Box decode + clip + validity (classes 1..20 flattened to 10240 entries)
// ---------------------------------------------------------------------------
__global__ void decode_kernel(const float* __restrict__ props,
                              const float* __restrict__ reg,
                              const float* __restrict__ probs,
                              const int* __restrict__ img_h_p,
                              const int* __restrict__ img_w_p,
                              float* __restrict__ boxes,
                              float* __restrict__ scores,
                              int* __restrict__ labels,
                              int* __restrict__ valid) {
  int i = blockIdx.x * blockDim.x + threadIdx.x;
  if (i >= NBOX) return;
  int n = i / 20;
  int c = (i % 20) + 1;
  float imw = (float)(*img_w_p);
  float imh = (float)(*img_h_p);

  float x1 = props[n * 4 + 0], y1 = props[n * 4 + 1];
  float x2 = props[n * 4 + 2], y2 = props[n * 4 + 3];
  float w = x2 - x1, h = y2 - y1;
  float cx = x1 + 0.5f * w, cy = y1 + 0.5f * h;

  const float* d = reg + (size_t)n * (NCLS * 4) + c * 4;
  float dx = d[0], dy = d[1];
  float dw = fminf(d[2], BBOX_CLIP);
  float dh = fminf(d[3], BBOX_CLIP);
  float pcx = dx * w + cx, pcy = dy * h + cy;
  float pw = expf(dw) * w, ph = expf(dh) * h;

  float bx1 = fminf(fmaxf(pcx - pw * 0.5f, 0.0f), imw);
  float by1 = fminf(fmaxf(pcy - ph * 0.5f, 0.0f), imh);
  float bx2 = fminf(fmaxf(pcx + pw * 0.5f, 0.0f), imw);
  float by2 = fminf(fmaxf(pcy + ph * 0.5f, 0.0f), imh);

  float sc = probs[(size_t)n * NCLS + c];
  boxes[i * 4 + 0] = bx1; boxes[i * 4 + 1] = by1;
  boxes[i * 4 + 2] = bx2; boxes[i * 4 + 3] = by2;
  scores[i] = sc;
  labels[i] = c;
  valid[i] = (sc > 0.05f) && ((bx2 - bx1) >= 1.0f) && ((by2 - by1) >= 1.0f);
}

// ---------------------------------------------------------------------------
// 5) Sort: init (key,idx) pairs, then bitonic passes (stable-equivalent:
//    descending score, ties broken by ascending index => JAX stable argsort)
// ---------------------------------------------------------------------------
__global__ void sort_init_kernel(const float* __restrict__ scores,
                                 float* __restrict__ keys,
                                 int* __restrict__ idxs) {
  int i = blockIdx.x * blockDim.x + threadIdx.x;
  if (i >= NSORT) return;
  keys[i] = (i < NBOX) ? scores[i] : -1e30f;
  idxs[i] = i;
}

__device__ __forceinline__ bool comes_before(float ka, int ia, float kb, int ib) {
  return (ka > kb) || (ka == kb && ia < ib);
}

__global__ void bitonic_step(float* __restrict__ keys, int* __restrict__ idxs,
                             int j, int k) {
  int i = blockIdx.x * blockDim.x + threadIdx.x;
  if (i >= NSORT) return;
  int ixj = i ^ j;
  if (ixj <= i) return;
  float ki = keys[i], kj = keys[ixj];
  int ii = idxs[i], ij = idxs[ixj];
  bool up = ((i & k) == 0);
  bool do_swap = up ? comes_before(kj, ij, ki, ii)
                    : comes_before(ki, ii, kj, ij);
  if (do_swap) {
    keys[i] = kj; keys[ixj] = ki;
    idxs[i] = ij; idxs[ixj] = ii;
  }
}

// ---------------------------------------------------------------------------
// 6) Gather sorted arrays; build NMS boxes (class offset) and their areas
// ---------------------------------------------------------------------------
__global__ void gather_kernel(const int* __restrict__ idxs,
                              const float* __restrict__ boxes,
                              const float* __restrict__ scores,
                              const int* __restrict__ labels,
                              const int* __restrict__ valid,
                              float* __restrict__ sb,     // sorted orig boxes
                              float* __restrict__ snb,    // + class offset
                              float* __restrict__ sarea,  // area of offset box
                              float* __restrict__ ss,
                              int* __restrict__ sl,
                              int* __restrict__ sv) {
  int s = blockIdx.x * blockDim.x + threadIdx.x;
  if (s >= NBOX) return;
  int i = idxs[s];
  float bx1 = boxes[i * 4 + 0], by1 = boxes[i * 4 + 1];
  float bx2 = boxes[i * 4 + 2], by2 = boxes[i * 4 + 3];
  int lb = labels[i];
  float off = 1000.0f * (float)lb;
  float nx1 = bx1 + off, ny1 = by1 + off, nx2 = bx2 + off, ny2 = by2 + off;
  sb[s * 4 + 0] = bx1; sb[s * 4 + 1] = by1; sb[s * 4 + 2] = bx2; sb[s * 4 + 3] = by2;
  snb[s * 4 + 0] = nx1; snb[s * 4 + 1] = ny1; snb[s * 4 + 2] = nx2; snb[s * 4 + 3] = ny2;
  sarea[s] = (nx2 - nx1) * (ny2 - ny1);
  ss[s] = scores[i];
  sl[s] = lb;
  sv[s] = valid[i];
}

// ---------------------------------------------------------------------------
// 7) NMS (reference fori_loop semantics) + emit first 100 kept into d_out
//    keep bitmask lives in LDS; boxes read through L2 (160 KB working set)
// ---------------------------------------------------------------------------
__global__ __launch_bounds__(1024) void nms_final_kernel(
    const float* __restrict__ snb, const float* __restrict__ sarea,
    const float* __restrict__ sb, const float* __restrict__ ss,
    const int* __restrict__ sl, const int* __restrict__ sv,
    float* __restrict__ out) {
  __shared__ unsigned keepw[NBOX / 32];
  const int tid = threadIdx.x;

  for (int w = tid; w < NBOX / 32; w += blockDim.x) {
    unsigned word = 0;
    for (int b = 0; b < 32; ++b)
      if (sv[w * 32 + b]) word |= (1u << b);
    keepw[w] = word;
  }
  __syncthreads();

  for (int i = 0; i < NBOX; ++i) {
    bool ki = (keepw[i >> 5] >> (i & 31)) & 1u;
    if (ki) {
      float ax1 = snb[i * 4 + 0], ay1 = snb[i * 4 + 1];
      float ax2 = snb[i * 4 + 2], ay2 = snb[i * 4 + 3];
      float aarea = sarea[i];
      for (int j = i + 1 + tid; j < NBOX; j += blockDim.x) {
        if (!((keepw[j >> 5] >> (j & 31)) & 1u)) continue;  // already gone
        float xl = fmaxf(ax1, snb[j * 4 + 0]);
        float yt = fmaxf(ay1, snb[j * 4 + 1]);
        float xr = fminf(ax2, snb[j * 4 + 2]);
        float yb = fminf(ay2, snb[j * 4 + 3]);
        float inter = fmaxf(xr - xl, 0.0f) * fmaxf(yb - yt, 0.0f);
        float iou = inter / (aarea + sarea[j] - inter);
        if (iou > 0.5f) atomicAnd(&keepw[j >> 5], ~(1u << (j & 31)));
      }
    }
    __syncthreads();
  }

  // zero the 600-element output, then serial emit of first TOPK kept
  if (tid < 600) out[tid] = 0.0f;
  __syncthreads();
  if (tid == 0) {
    int cnt = 0;
    for (int i = 0; i < NBOX && cnt < TOPK; ++i) {
      if ((keepw[i >> 5] >> (i & 31)) & 1u) {
        out[cnt * 4 + 0] = sb[i * 4 + 0];
        out[cnt * 4 + 1] = sb[i * 4 + 1];
        out[cnt * 4 + 2] = sb[i * 4 + 2];
        out[cnt * 4 + 3] = sb[i * 4 + 3];
        out[400 + cnt] = ss[i];
        out[500 + cnt] = (float)sl[i];
        ++cnt;
      }
    }
  }
}

// ---------------------------------------------------------------------------
// Launch
// ---------------------------------------------------------------------------
extern "C" void kernel_launch(void* const* d_in, const int* in_sizes, int n_in,
                              void* d_out, int out_size, void* d_ws, size_t ws_size,
                              hipStream_t stream) {
  const float* feat  = (const float*)d_in[0];
  const float* props = (const float*)d_in[1];
  const float* w6 = (const float*)d_in[2];
  const float* b6 = (const float*)d_in[3];
  const float* w7 = (const float*)d_in[4];
  const float* b7 = (const float*)d_in[5];
  const float* wc = (const float*)d_in[6];
  const float* bc = (const float*)d_in[7];
  const float* wr = (const float*)d_in[8];
  const float* br = (const float*)d_in[9];
  const int* img_h = (const int*)d_in[10];
  const int* img_w = (const int*)d_in[11];
  float* out = (float*)d_out;

  // workspace carve-out (256B aligned)
  uint8_t* p = (uint8_t*)d_ws;
  size_t off = 0;
  auto alloc = [&](size_t bytes) -> void* {
    void* r = p + off;
    off += (bytes + 255) & ~(size_t)255;
    return r;
  };
  float* X     = (float*)alloc((size_t)NROI * DIN * 4);
  float* H6    = (float*)alloc((size_t)NROI * DHID * 4);
  float* H7    = (float*)alloc((size_t)NROI * DHID * 4);
  float* CLS   = (float*)alloc((size_t)NROI * NCLS * 4);
  float* PROBS = (float*)alloc((size_t)NROI * NCLS * 4);
  float* REG   = (float*)alloc((size_t)NROI * NCLS * 4 * 4);
  float* BOX   = (float*)alloc((size_t)NBOX * 4 * 4);
  float* SC    = (float*)alloc((size_t)NBOX * 4);
  int*   LB    = (int*)  alloc((size_t)NBOX * 4);
  int*   VD    = (int*)  alloc((size_t)NBOX * 4);
  float* KEY   = (float*)alloc((size_t)NSORT * 4);
  int*   IDX   = (int*)  alloc((size_t)NSORT * 4);
  float* SB    = (float*)alloc((size_t)NBOX * 4 * 4);
  float* SNB   = (float*)alloc((size_t)NBOX * 4 * 4);
  float* SAR   = (float*)alloc((size_t)NBOX * 4);
  float* SS    = (float*)alloc((size_t)NBOX * 4);
  int*   SL    = (int*)  alloc((size_t)NBOX * 4);
  int*   SV    = (int*)  alloc((size_t)NBOX * 4);
  (void)ws_size; (void)in_sizes; (void)n_in; (void)out_size;

  // 1) ROI pool
  {
    int total = NROI * NCH * 49;
    roi_pool_kernel<<<(total + 255) / 256, 256, 0, stream>>>(feat, props, X);
  }
  // 2) FC6: [512,12544] x [1024,12544]^T -> relu
  gemm_wmma_f32<<<dim3(NROI / 64, DHID / 64), 128, 0, stream>>>(
      X, w6, b6, H6, DIN, DHID, 1);
  // 3) FC7: [512,1024] x [1024,1024]^T -> relu
  gemm_wmma_f32<<<dim3(NROI / 64, DHID / 64), 128, 0, stream>>>(
      H6, w7, b7, H7, DHID, DHID, 1);
  // 4) cls head: N=21
  gemm_wmma_f32<<<dim3(NROI / 64, 1), 128, 0, stream>>>(
      H7, wc, bc, CLS, DHID, NCLS, 0);
  // 5) reg head: N=84
  gemm_wmma_f32<<<dim3(NROI / 64, 2), 128, 0, stream>>>(
      H7, wr, br, REG, DHID, NCLS * 4, 0);
  // 6) softmax
  softmax_kernel<<<(NROI + 255) / 256, 256, 0, stream>>>(CLS, PROBS);
  // 7) decode
  decode_kernel<<<(NBOX + 255) / 256, 256, 0, stream>>>(
      props, REG, PROBS, img_h, img_w, BOX, SC, LB, VD);
  // 8) sort (stable-equivalent bitonic over 16384)
  sort_init_kernel<<<NSORT / 256, 256, 0, stream>>>(SC, KEY, IDX);
  for (int k = 2; k <= NSORT; k <<= 1)
    for (int j = k >> 1; j > 0; j >>= 1)
      bitonic_step<<<NSORT / 256, 256, 0, stream>>>(KEY, IDX, j, k);
  // 9) gather sorted
  gather_kernel<<<(NBOX + 255) / 256, 256, 0, stream>>>(
      IDX, BOX, SC, LB, VD, SB, SNB, SAR, SS, SL, SV);
  // 10) NMS + top-100 emit
  nms_final_kernel<<<1, 1024, 0, stream>>>(SNB, SAR, SB, SS, SL, SV, out);
}